// CTProjector3DModule_50955491999787
// MI455X (gfx1250) — compile-verified
//
#include <hip/hip_runtime.h>
#include <hip/hip_bf16.h>
#include <math.h>

// CT forward projection: out[r] = sum_k inb * volume[floor(M*p+b)] * (t1-t0)*|dst-src|
// N=256 volume, R=65536 rays, K=512 samples (511 segments) per ray.
// Strategy: 1 wave32 per ray. Async-stage the ray's 2KB tvals row into LDS
// (gfx1250 GLOBAL_LOAD_ASYNC_TO_LDS_B128, non-temporal so the 128MB stream
// doesn't evict the L2-resident 64MB volume), then each lane handles 16
// segments, gathers from the volume, and the wave reduces via shuffles.

#define N_DIM  256
#define K_SAMP 512
#define WAVES_PER_BLOCK 8

#if defined(__has_builtin)
#  if __has_builtin(__builtin_amdgcn_global_load_async_to_lds_b128)
#    define HAVE_ASYNC_B128 1
#  endif
#  if __has_builtin(__builtin_amdgcn_s_wait_asynccnt)
#    define HAVE_WAIT_ASYNC 1
#  endif
#endif

typedef int   v4i __attribute__((ext_vector_type(4)));
typedef float v4f __attribute__((ext_vector_type(4)));
typedef __attribute__((address_space(1))) v4i* as1_v4i_ptr;   // global
typedef __attribute__((address_space(3))) v4i* as3_v4i_ptr;   // LDS

__device__ __forceinline__ void wait_asynccnt_zero() {
#if defined(HAVE_WAIT_ASYNC)
    __builtin_amdgcn_s_wait_asynccnt(0);
#elif defined(HAVE_ASYNC_B128)
    asm volatile("s_wait_asynccnt 0" ::: "memory");
#endif
}

__global__ __launch_bounds__(WAVES_PER_BLOCK * 32) void ct_forward_project_kernel(
    const float* __restrict__ volume,   // [256*256*256]
    const float* __restrict__ tvals,    // [R, 512]
    const float* __restrict__ src,      // [R, 3]
    const float* __restrict__ dst,      // [R, 3]
    const float* __restrict__ Mmat,     // [3,3]
    const float* __restrict__ bvec,     // [3]
    float* __restrict__ out,            // [R]
    int R)
{
    __shared__ float lds_t[WAVES_PER_BLOCK * K_SAMP];   // 16 KB

    const int lane = threadIdx.x & 31;
    const int wave = threadIdx.x >> 5;
    const int ray  = blockIdx.x * WAVES_PER_BLOCK + wave;
    if (ray >= R) return;   // uniform per wave

    const float* row  = tvals + (size_t)ray * K_SAMP;
    float*       wlds = &lds_t[wave * K_SAMP];

    // ---- Stage 512 floats (2 KB) of tvals into LDS: 4 x B128 per lane ----
#if defined(HAVE_ASYNC_B128)
    #pragma unroll
    for (int it = 0; it < 4; ++it) {
        const int slot = it * 32 + lane;          // 128 slots x 16B = 2KB
        __builtin_amdgcn_global_load_async_to_lds_b128(
            (as1_v4i_ptr)(row  + slot * 4),
            (as3_v4i_ptr)(wlds + slot * 4),
            /*offset=*/0, /*cpol: TH_NT=1 (don't pollute L2; keep volume resident)*/1);
    }
    wait_asynccnt_zero();
#else
    #pragma unroll
    for (int it = 0; it < 4; ++it) {
        const int slot = it * 32 + lane;
        const v4f v = __builtin_nontemporal_load((const v4f*)(row + slot * 4));
        *(v4f*)(wlds + slot * 4) = v;
    }
#endif
    // Wave-private LDS region; asynccnt/dscnt ordering covers write->read.

    // ---- Per-ray geometry (uniform across the wave; cache-broadcast) ----
    const float sx = src[3 * ray + 0];
    const float sy = src[3 * ray + 1];
    const float sz = src[3 * ray + 2];
    const float dx = dst[3 * ray + 0] - sx;
    const float dy = dst[3 * ray + 1] - sy;
    const float dz = dst[3 * ray + 2] - sz;
    const float ray_len = sqrtf(dx * dx + dy * dy + dz * dz);

    const float m00 = Mmat[0], m01 = Mmat[1], m02 = Mmat[2];
    const float m10 = Mmat[3], m11 = Mmat[4], m12 = Mmat[5];
    const float m20 = Mmat[6], m21 = Mmat[7], m22 = Mmat[8];
    const float b0 = bvec[0], b1 = bvec[1], b2 = bvec[2];

    float acc = 0.0f;

    // ---- 511 segments, strided by 32 lanes, 16 rounds ----
    #pragma unroll
    for (int j = 0; j < K_SAMP / 32; ++j) {
        const int k = j * 32 + lane;
        if (k < K_SAMP - 1) {
            const float t0 = wlds[k];
            const float t1 = wlds[k + 1];
            // tvals are in (0,1) or +inf; finite check:
            if (t0 < 1.0e30f && t1 < 1.0e30f) {
                const float seg  = (t1 - t0) * ray_len;
                const float tmid = 0.5f * (t0 + t1);
                const float px = fmaf(tmid, dx, sx);
                const float py = fmaf(tmid, dy, sy);
                const float pz = fmaf(tmid, dz, sz);
                const float qx = fmaf(m00, px, fmaf(m01, py, fmaf(m02, pz, b0)));
                const float qy = fmaf(m10, px, fmaf(m11, py, fmaf(m12, pz, b1)));
                const float qz = fmaf(m20, px, fmaf(m21, py, fmaf(m22, pz, b2)));
                const int i0 = (int)floorf(qx);
                const int i1 = (int)floorf(qy);
                const int i2 = (int)floorf(qz);
                if (((unsigned)i0 < (unsigned)N_DIM) &
                    ((unsigned)i1 < (unsigned)N_DIM) &
                    ((unsigned)i2 < (unsigned)N_DIM)) {
                    const float v = volume[((size_t)i0 * N_DIM + i1) * N_DIM + i2];
                    acc = fmaf(v, seg, acc);
                }
            }
        }
    }

    // ---- wave32 reduction ----
    #pragma unroll
    for (int off = 16; off > 0; off >>= 1)
        acc += __shfl_down(acc, off, 32);

    if (lane == 0) out[ray] = acc;
}

extern "C" void kernel_launch(void* const* d_in, const int* in_sizes, int n_in,
                              void* d_out, int out_size, void* d_ws, size_t ws_size,
                              hipStream_t stream) {
    const float* volume = (const float*)d_in[0];
    const float* tvals  = (const float*)d_in[1];
    const float* src    = (const float*)d_in[2];
    const float* dst    = (const float*)d_in[3];
    const float* Mmat   = (const float*)d_in[4];
    const float* bvec   = (const float*)d_in[5];
    float* out = (float*)d_out;

    const int R = in_sizes[1] / K_SAMP;            // 65536
    const int blocks = (R + WAVES_PER_BLOCK - 1) / WAVES_PER_BLOCK;   // 8192
    ct_forward_project_kernel<<<blocks, WAVES_PER_BLOCK * 32, 0, stream>>>(
        volume, tvals, src, dst, Mmat, bvec, out, R);
}